// SpatialTemporalGNN_Hidden_40252433498166
// MI455X (gfx1250) — compile-verified
//
#include <hip/hip_runtime.h>
#include <hip/hip_bf16.h>

typedef __attribute__((ext_vector_type(2))) float v2f;
typedef __attribute__((ext_vector_type(8))) float v8f;

#define NPG 62          // nodes per graph
#define EPG 496         // edges per graph
#define CIN 32
#define COUT 64
#define NGRAPH 2048
#define EDGE_TOT (NGRAPH * EPG)
#define GRU_H 128
#define SEQ 64
#define BATCH 32

// order-preserving float<->int encoding for LDS atomicMax
__device__ __forceinline__ int enc_f(float f) {
    int i = __float_as_int(f);
    return (i < 0) ? (i ^ 0x7FFFFFFF) : i;
}
__device__ __forceinline__ float dec_f(int i) {
    return __int_as_float((i < 0) ? (i ^ 0x7FFFFFFF) : i);
}

// ---------------------------------------------------------------------------
// Kernel 1: fused GATv2Conv + mean pool. One block per graph, 256 thr = 8 waves.
// ---------------------------------------------------------------------------
__global__ __launch_bounds__(256) void gat_pool_kernel(
    const float* __restrict__ x,
    const long long* __restrict__ eidx,   // [2, E] int64
    const float* __restrict__ eattr,      // [E, 1]
    const float* __restrict__ Wl, const float* __restrict__ bl,
    const float* __restrict__ Wr, const float* __restrict__ br,
    const float* __restrict__ We,         // [64,1]
    const float* __restrict__ att,        // [64]
    const float* __restrict__ bias,       // [64]
    float* __restrict__ xg)               // [NGRAPH, 64]
{
    __shared__ float xl_s[64 * 65];
    __shared__ float xr_s[64 * 65];
    __shared__ float buf_s[64 * 65];      // phase 1: x tile (stride 33); phase 2: accumulator (stride 65)
    __shared__ float deg_s[64];
    __shared__ float la_s[64];
    __shared__ float asum_s[64];
    __shared__ int   amax_s[64];
    __shared__ int   esrc_s[EPG];
    __shared__ int   edst_s[EPG];
    __shared__ float ea_s[EPG];
    __shared__ float a_s[EPG + 64];
    __shared__ float att_s[64];
    __shared__ float we_s[64];

    const int tid   = threadIdx.x;
    const int g     = blockIdx.x;
    const int nbase = g * NPG;

    // ---- load x tile (padded to 64 rows, stride 33) + small vectors ----
    for (int i = tid; i < 64 * CIN; i += 256) {
        int v = i >> 5, c = i & 31;
        buf_s[v * 33 + c] = (v < NPG) ? x[(size_t)(nbase + v) * CIN + c] : 0.0f;
    }
    if (tid < 64) {
        att_s[tid]  = att[tid];
        we_s[tid]   = We[tid];
        deg_s[tid]  = 0.0f;
        la_s[tid]   = 0.0f;
        asum_s[tid] = 0.0f;
        amax_s[tid] = (int)0x80000000;    // -inf under enc ordering
    }
    __syncthreads();

    // ---- xl = x @ Wl^T + bl ; xr = x @ Wr^T + br via V_WMMA_F32_16X16X4_F32 ----
    // Output 64x64 per matrix = 4x4 tiles of 16x16; 8 waves * 2 tiles each.
    const int lane = tid & 31;
    const int wv   = tid >> 5;
    const int hf   = lane >> 4;           // half: 0 -> lanes 0-15, 1 -> lanes 16-31
    const int mrow = lane & 15;

    for (int mat = 0; mat < 2; ++mat) {
        const float* W    = mat ? Wr : Wl;
        const float* bvec = mat ? br : bl;
        float*       dst  = mat ? xr_s : xl_s;
        for (int tt = wv; tt < 16; tt += 8) {     // uniform across wave: EXEC all 1s
            const int mi = tt >> 2, ni = tt & 3;
            v8f acc = {};
            const int arow = mi * 16 + mrow;
            const int n    = ni * 16 + mrow;
            #pragma unroll
            for (int ks = 0; ks < 8; ++ks) {
                const int k0 = ks * 4 + 2 * hf;
                v2f a, b;
                a.x = buf_s[arow * 33 + k0];
                a.y = buf_s[arow * 33 + k0 + 1];
                b.x = W[n * CIN + k0];            // B[k][n] = W^T[k][n] = W[n][k]
                b.y = W[n * CIN + k0 + 1];
                acc = __builtin_amdgcn_wmma_f32_16x16x4_f32(
                        false, a, false, b, (short)0, acc, false, false);
            }
            const float bv = bvec[n];
            #pragma unroll
            for (int r = 0; r < 8; ++r) {
                const int row = mi * 16 + 8 * hf + r;   // C/D layout: M = r + 8*half
                dst[row * 65 + n] = acc[r] + bv;
            }
        }
    }
    __syncthreads();

    // ---- zero accumulator; load edges; deg & loop_attr numerator ----
    for (int i = tid; i < 64 * 65; i += 256) buf_s[i] = 0.0f;
    for (int e = tid; e < EPG; e += 256) {
        const size_t eg = (size_t)g * EPG + e;
        const int s = (int)eidx[eg] - nbase;
        const int d = (int)eidx[(size_t)EDGE_TOT + eg] - nbase;
        const float w = eattr[eg];
        esrc_s[e] = s; edst_s[e] = d; ea_s[e] = w;
        atomicAdd(&deg_s[d], 1.0f);
        atomicAdd(&la_s[d], w);
    }
    __syncthreads();
    if (tid < NPG) la_s[tid] = la_s[tid] / fmaxf(deg_s[tid], 1.0f);
    __syncthreads();

    // ---- attention logits a = leaky_relu(xl[src]+xr[dst]+ea*We, 0.2) . att ----
    for (int e = tid; e < EPG + NPG; e += 256) {
        int s, d; float w;
        if (e < EPG) { s = esrc_s[e]; d = edst_s[e]; w = ea_s[e]; }
        else         { s = d = e - EPG; w = la_s[s]; }          // self-loop, mean attr
        float acc = 0.0f;
        for (int c = 0; c < COUT; ++c) {
            const float ev = xl_s[s * 65 + c] + xr_s[d * 65 + c] + w * we_s[c];
            const float lr = ev > 0.0f ? ev : 0.2f * ev;
            acc += lr * att_s[c];
        }
        a_s[e] = acc;
        atomicMax(&amax_s[d], enc_f(acc));
    }
    __syncthreads();

    // ---- softmax numerators + denominators ----
    for (int e = tid; e < EPG + NPG; e += 256) {
        const int d = (e < EPG) ? edst_s[e] : (e - EPG);
        const float ae = expf(a_s[e] - dec_f(amax_s[d]));
        a_s[e] = ae;
        atomicAdd(&asum_s[d], ae);
    }
    __syncthreads();

    // ---- weighted aggregation: acc[dst] += alpha * xl[src] ----
    for (int e = tid; e < EPG + NPG; e += 256) {
        int s, d;
        if (e < EPG) { s = esrc_s[e]; d = edst_s[e]; }
        else         { s = d = e - EPG; }
        const float alpha = a_s[e] / asum_s[d];
        for (int c = 0; c < COUT; ++c)
            atomicAdd(&buf_s[d * 65 + c], alpha * xl_s[s * 65 + c]);
    }
    __syncthreads();

    // ---- mean pool over 62 nodes (+ output bias) ----
    if (tid < COUT) {
        float sum = 0.0f;
        for (int v = 0; v < NPG; ++v) sum += buf_s[v * 65 + tid];
        xg[(size_t)g * COUT + tid] = sum * (1.0f / (float)NPG) + bias[tid];
    }
}

// ---------------------------------------------------------------------------
// Kernel 2: gx = x_graph @ Wih^T + bih   ([2048,64] x [64,384]) via WMMA f32.
// One 16x16 output tile per wave; 128*24 = 3072 tiles; 8 waves/block -> 384 blocks.
// ---------------------------------------------------------------------------
__global__ __launch_bounds__(256) void gru_gx_kernel(
    const float* __restrict__ xg,
    const float* __restrict__ Wih,       // [384, 64]
    const float* __restrict__ bih,       // [384]
    float* __restrict__ gx)              // [2048, 384]
{
    const int wid  = blockIdx.x * 8 + (threadIdx.x >> 5);
    const int lane = threadIdx.x & 31;
    const int hf   = lane >> 4;
    const int mrow = lane & 15;
    const int mi = wid / 24, ni = wid % 24;

    v8f acc = {};
    const int arow = (mi * 16 + mrow) * 64;
    const int brow = (ni * 16 + mrow) * 64;
    #pragma unroll
    for (int ks = 0; ks < 16; ++ks) {
        const int k0 = ks * 4 + 2 * hf;
        v2f a, b;
        a.x = xg[arow + k0];  a.y = xg[arow + k0 + 1];
        b.x = Wih[brow + k0]; b.y = Wih[brow + k0 + 1];
        acc = __builtin_amdgcn_wmma_f32_16x16x4_f32(
                false, a, false, b, (short)0, acc, false, false);
    }
    const int n = ni * 16 + mrow;
    const float bv = bih[n];
    #pragma unroll
    for (int r = 0; r < 8; ++r) {
        const int m = mi * 16 + 8 * hf + r;
        gx[(size_t)m * 384 + n] = acc[r] + bv;
    }
}

// ---------------------------------------------------------------------------
// Kernel 3: GRU recurrence. One block per batch row (32 independent sequences),
// 384 threads: thread i owns gate row i of Whh ([384,128], L2-resident).
// ---------------------------------------------------------------------------
__global__ __launch_bounds__(384) void gru_seq_kernel(
    const float* __restrict__ gx,        // [2048, 384]  (b*64+t rows)
    const float* __restrict__ Whh,       // [384, 128]
    const float* __restrict__ bhh,       // [384]
    float* __restrict__ hT)              // [32, 128]
{
    __shared__ float h_s[GRU_H];
    __shared__ float gh_s[3 * GRU_H];

    const int tid = threadIdx.x;
    const int b   = blockIdx.x;
    if (tid < GRU_H) h_s[tid] = 0.0f;
    __syncthreads();

    const float* wrow = Whh + (size_t)tid * GRU_H;
    const float  bh   = bhh[tid];

    for (int t = 0; t < SEQ; ++t) {
        float acc = bh;
        #pragma unroll 4
        for (int k = 0; k < GRU_H; k += 4) {
            acc += wrow[k]     * h_s[k]
                 + wrow[k + 1] * h_s[k + 1]
                 + wrow[k + 2] * h_s[k + 2]
                 + wrow[k + 3] * h_s[k + 3];
        }
        gh_s[tid] = acc;
        __syncthreads();
        if (tid < GRU_H) {
            const float* gxt = gx + (size_t)(b * SEQ + t) * 384;
            const float r  = 1.0f / (1.0f + expf(-(gxt[tid]           + gh_s[tid])));
            const float z  = 1.0f / (1.0f + expf(-(gxt[GRU_H + tid]   + gh_s[GRU_H + tid])));
            const float nn = tanhf(gxt[2 * GRU_H + tid] + r * gh_s[2 * GRU_H + tid]);
            h_s[tid] = (1.0f - z) * nn + z * h_s[tid];
        }
        __syncthreads();
    }
    if (tid < GRU_H) hT[(size_t)b * GRU_H + tid] = h_s[tid];
}

// ---------------------------------------------------------------------------
// Kernel 4: classifier  relu(hT @ W1^T + b1) @ W2^T + b2  -> [32, 3]
// ---------------------------------------------------------------------------
__global__ __launch_bounds__(256) void cls_kernel(
    const float* __restrict__ hT,
    const float* __restrict__ W1, const float* __restrict__ b1,  // [64,128],[64]
    const float* __restrict__ W2, const float* __restrict__ b2,  // [3,64],[3]
    float* __restrict__ out)                                     // [32,3]
{
    __shared__ float hid_s[BATCH * 64];
    const int tid = threadIdx.x;
    for (int i = tid; i < BATCH * 64; i += 256) {
        const int bb = i >> 6, j = i & 63;
        const float* h = hT + (size_t)bb * GRU_H;
        const float* w = W1 + (size_t)j * GRU_H;
        float acc = b1[j];
        for (int k = 0; k < GRU_H; ++k) acc += w[k] * h[k];
        hid_s[i] = acc > 0.0f ? acc : 0.0f;
    }
    __syncthreads();
    for (int i = tid; i < BATCH * 3; i += 256) {
        const int bb = i / 3, k = i % 3;
        const float* hv = hid_s + bb * 64;
        const float* w  = W2 + (size_t)k * 64;
        float acc = b2[k];
        for (int c = 0; c < 64; ++c) acc += w[c] * hv[c];
        out[i] = acc;
    }
}

extern "C" void kernel_launch(void* const* d_in, const int* in_sizes, int n_in,
                              void* d_out, int out_size, void* d_ws, size_t ws_size,
                              hipStream_t stream) {
    const float*     x     = (const float*)d_in[0];
    const long long* eidx  = (const long long*)d_in[1];   // int64 [2, E]
    const float*     eattr = (const float*)d_in[2];
    const float*     Wl    = (const float*)d_in[3];
    const float*     bl    = (const float*)d_in[4];
    const float*     Wr    = (const float*)d_in[5];
    const float*     br    = (const float*)d_in[6];
    const float*     We    = (const float*)d_in[7];
    const float*     att   = (const float*)d_in[8];
    const float*     bias  = (const float*)d_in[9];
    const float*     Wih   = (const float*)d_in[10];
    const float*     Whh   = (const float*)d_in[11];
    const float*     bih   = (const float*)d_in[12];
    const float*     bhh   = (const float*)d_in[13];
    const float*     W1    = (const float*)d_in[14];
    const float*     b1    = (const float*)d_in[15];
    const float*     W2    = (const float*)d_in[16];
    const float*     b2    = (const float*)d_in[17];

    float* ws  = (float*)d_ws;
    float* xg  = ws;                                 // 2048*64
    float* gxb = xg + (size_t)NGRAPH * COUT;         // 2048*384
    float* hT  = gxb + (size_t)NGRAPH * 3 * GRU_H;   // 32*128
    float* out = (float*)d_out;

    gat_pool_kernel<<<NGRAPH, 256, 0, stream>>>(x, eidx, eattr, Wl, bl, Wr, br,
                                                We, att, bias, xg);
    gru_gx_kernel<<<384, 256, 0, stream>>>(xg, Wih, bih, gxb);
    gru_seq_kernel<<<BATCH, 384, 0, stream>>>(gxb, Whh, bhh, hT);
    cls_kernel<<<1, 256, 0, stream>>>(hT, W1, b1, W2, b2, out);
}